// HybridModel_79010218377532
// MI455X (gfx1250) — compile-verified
//
#include <hip/hip_runtime.h>
#include <hip/hip_bf16.h>
#include <math.h>
#include <stdint.h>

// ---------------------------------------------------------------------------
// Hybrid quantum-classical model, MI455X (gfx1250, wave32).
//   Stage 1: 2560 workgroups, each simulates one 10-qubit circuit with the
//            1024-amplitude state resident in LDS (re/im split, 8 KB).
//            Per-circuit weights staged via global_load_async_to_lds_b32.
//   Stage 2: single-wave WMMA (v_wmma_f32_16x16x4_f32) head: z @ W2^T + b2,
//            softmax, f32 throughout. Unconditional masked loads (no exec
//            branches around VMEM).
// ---------------------------------------------------------------------------

#define NCIRC 320
#define NB    8
#define DIM   1024

typedef __attribute__((ext_vector_type(2))) float v2f;
typedef __attribute__((ext_vector_type(8))) float v8f;

// Apply a general complex 2x2 gate U to qubit q (bit position 9-q).
// 512 pairs, 2 per thread. Caller guarantees uniform control flow.
__device__ __forceinline__ void gate1q(float* re, float* im, int q,
    float u00r, float u00i, float u01r, float u01i,
    float u10r, float u10i, float u11r, float u11i) {
  const int s = 1 << (9 - q);
#pragma unroll
  for (int it = 0; it < 2; ++it) {
    int k  = threadIdx.x + it * 256;
    int i0 = ((k & ~(s - 1)) << 1) | (k & (s - 1));
    int i1 = i0 + s;
    float ar = re[i0], ai = im[i0];
    float br = re[i1], bi = im[i1];
    re[i0] = u00r * ar - u00i * ai + u01r * br - u01i * bi;
    im[i0] = u00r * ai + u00i * ar + u01r * bi + u01i * br;
    re[i1] = u10r * ar - u10i * ai + u11r * br - u11i * bi;
    im[i1] = u10r * ai + u10i * ar + u11r * bi + u11i * br;
  }
  __syncthreads();
}

// Controlled 2x2 gate: control qubit j (bit 9-j), target j+1 (bit bt=8-j).
// Only the control=1 subspace: 256 pairs, one per thread.
__device__ __forceinline__ void gatec(float* re, float* im, int j,
    float u00r, float u00i, float u01r, float u01i,
    float u10r, float u10i, float u11r, float u11i) {
  const int bt = 8 - j;
  const int s  = 1 << bt;
  int k  = threadIdx.x;
  int i0 = ((k >> bt) << (bt + 2)) | (s << 1) | (k & (s - 1));
  int i1 = i0 + s;
  float ar = re[i0], ai = im[i0];
  float br = re[i1], bi = im[i1];
  re[i0] = u00r * ar - u00i * ai + u01r * br - u01i * bi;
  im[i0] = u00r * ai + u00i * ar + u01r * bi + u01i * br;
  re[i1] = u10r * ar - u10i * ai + u11r * br - u11i * bi;
  im[i1] = u10r * ai + u10i * ar + u11r * bi + u11i * br;
  __syncthreads();
}

__global__ __launch_bounds__(256) void qsim_kernel(
    const float* __restrict__ x, const float* __restrict__ W1,
    const float* __restrict__ b1, const float* __restrict__ qw,
    float* __restrict__ z) {
  __shared__ float re[DIM], im[DIM];
  __shared__ float gm[65][8];    // 60 fused-Rot matrices + 5 embedding gates
  __shared__ float qwlds[152];   // this circuit's 150 qweight angles
  __shared__ float red[8];       // per-wave partials

  const int inst = blockIdx.x;          // b*320 + c
  const int b    = inst / NCIRC;
  const int c    = inst - b * NCIRC;
  const int tid  = threadIdx.x;

  // Stage this circuit's 150 angles into LDS with the gfx1250 async path
  // (ASYNCcnt-tracked, overlaps with the state init below).
  if (tid < 150) {
    unsigned lds_off = (unsigned)(size_t)(&qwlds[tid]);  // low 32b = LDS offset
    unsigned goff    = (unsigned)tid * 4u;
    const float* base = qw + (size_t)c * 150;
    asm volatile("global_load_async_to_lds_b32 %0, %1, %2"
                 :: "v"(lds_off), "v"(goff), "s"(base) : "memory");
  }

  // |0..0> followed by H on every qubit == uniform real state 1/32.
  for (int i = tid; i < DIM; i += 256) { re[i] = 0.03125f; im[i] = 0.0f; }

  asm volatile("s_wait_asynccnt 0x0" ::: "memory");
  __syncthreads();

  if (tid < 60) {
    // Fused Rot(a,bb,cc) = RZ(cc)*RY(bb)*RZ(a). Pass 5 reuses layer-4 weights
    // (reference applies w[:, -1] again after the layer loop).
    int l = tid / 10, q = tid - l * 10;
    int wl = (l == 5) ? 4 : l;
    const float* a3 = &qwlds[(wl * 10 + q) * 3];
    float a = a3[0], bb = a3[1], cc = a3[2];
    float cb = cosf(0.5f * bb), sb = sinf(0.5f * bb);
    float ap = 0.5f * (a + cc), am = 0.5f * (a - cc);
    float cap = cosf(ap), sap = sinf(ap);
    float cam = cosf(am), sam = sinf(am);
    float* g = gm[tid];
    g[0] =  cb * cap;  g[1] = -cb * sap;    // U00 = cb e^{-i(a+c)/2}
    g[2] = -sb * cam;  g[3] = -sb * sam;    // U01 = -sb e^{+i(a-c)/2}
    g[4] =  sb * cam;  g[5] = -sb * sam;    // U10 =  sb e^{-i(a-c)/2}
    g[6] =  cb * cap;  g[7] =  cb * sap;    // U11 = cb e^{+i(a+c)/2}
  } else if (tid < 65) {
    // Embedding gate for layer l: RY(h) (even l) or RZ(h) (odd l),
    // h = x[b] . W1[c] + b1[c].
    int l = tid - 60;
    float h  = x[b * 2 + 0] * W1[c * 2 + 0] + x[b * 2 + 1] * W1[c * 2 + 1] + b1[c];
    float ch = cosf(0.5f * h), sh = sinf(0.5f * h);
    float* g = gm[tid];
    if ((l & 1) == 0) { g[0]=ch; g[1]=0; g[2]=-sh; g[3]=0; g[4]=sh; g[5]=0; g[6]=ch; g[7]=0; }
    else              { g[0]=ch; g[1]=-sh; g[2]=0; g[3]=0; g[4]=0;  g[5]=0; g[6]=ch; g[7]=sh; }
  }
  __syncthreads();

  const float R2 = 0.70710678118654752f;   // 1/sqrt(2)
  const float CQ = 0.92387953251128674f;   // cos(pi/8)
  const float SQ = 0.38268343236508977f;   // sin(pi/8)
  // Fused CRY(pi/4) * CRZ(pi/2) * CNOT acting on the control=1 subspace:
  //   U = [[-sq e^{+i pi/4}, cq e^{-i pi/4}], [cq e^{+i pi/4}, sq e^{-i pi/4}]]
  const float f00r = -SQ * R2, f00i = -SQ * R2;
  const float f01r =  CQ * R2, f01i = -CQ * R2;
  const float f10r =  CQ * R2, f10i =  CQ * R2;
  const float f11r =  SQ * R2, f11i = -SQ * R2;

  for (int l = 0; l < 5; ++l) {
    { const float* g = gm[60 + l];
      gate1q(re, im, 0, g[0],g[1],g[2],g[3],g[4],g[5],g[6],g[7]); }
    for (int j = 0; j < 9; ++j)
      gatec(re, im, j, f00r,f00i,f01r,f01i,f10r,f10i,f11r,f11i);
    for (int q = 0; q < 10; ++q) {
      const float* g = gm[l * 10 + q];
      gate1q(re, im, q, g[0],g[1],g[2],g[3],g[4],g[5],g[6],g[7]);
    }
  }
  for (int q = 0; q < 10; ++q) {          // final Rot pass (w[:, -1])
    const float* g = gm[50 + q];
    gate1q(re, im, q, g[0],g[1],g[2],g[3],g[4],g[5],g[6],g[7]);
  }
  for (int q = 0; q < 10; ++q)            // final H wall
    gate1q(re, im, q, R2,0.f, R2,0.f, R2,0.f, -R2,0.f);

  // <Z_0>: +|amp|^2 for bit9==0, -|amp|^2 for bit9==1.
  float part = 0.f;
#pragma unroll
  for (int it = 0; it < 4; ++it) {
    int i = tid + it * 256;
    float v = re[i] * re[i] + im[i] * im[i];
    part += (i & 512) ? -v : v;
  }
  // wave32 reduction, then one cross-wave pass (1 barrier instead of 8).
#pragma unroll
  for (int off = 16; off > 0; off >>= 1) part += __shfl_down(part, off, 32);
  if ((tid & 31) == 0) red[tid >> 5] = part;
  __syncthreads();
  if (tid == 0) {
    float s = 0.f;
#pragma unroll
    for (int w = 0; w < 8; ++w) s += red[w];
    z[inst] = s;
  }
}

// Head: logits = z(8x320) @ W2^T(320x2) + b2, softmax over 2 classes.
// Single wave32, V_WMMA_F32_16X16X4_F32, K stepped by 4 (80 iterations).
// A frag (16x4 f32): lanes 0-15 hold M=lane, {K=k0,k0+1}; lanes 16-31 {K=k0+2,k0+3}.
// B frag (4x16 f32): assumed to mirror A's half-wave K split: half g holds
// rows K=2g,2g+1, column = lane&15 (not hardware-verified; compile-only env).
// Loads are unconditional (clamped row + 0/1 mask) to avoid exec-mask branches.
__global__ __launch_bounds__(32) void head_kernel(
    const float* __restrict__ z, const float* __restrict__ W2,
    const float* __restrict__ b2, float* __restrict__ out) {
  __shared__ float lg[8][2];
  const int lane  = threadIdx.x & 31;
  const int m     = lane & 15;
  const int g     = lane >> 4;
  const float mA  = (m < 8) ? 1.f : 0.f;
  const float mB  = (m < 2) ? 1.f : 0.f;
  const int rowA  = (m & 7) * 320;
  const int rowB  = (m & 1) * 320;

  v8f acc = {};
  for (int k0 = 0; k0 < 320; k0 += 4) {
    const int ka = k0 + 2 * g;
    v2f a, bm;
    a.x  = z[rowA + ka]      * mA;    // A[m][ka], A[m][ka+1]
    a.y  = z[rowA + ka + 1]  * mA;
    bm.x = W2[rowB + ka]     * mB;    // B[ka][n] = W2[n][ka], n = m
    bm.y = W2[rowB + ka + 1] * mB;
    acc = __builtin_amdgcn_wmma_f32_16x16x4_f32(
        /*neg_a=*/false, a, /*neg_b=*/false, bm,
        /*c_mod=*/(short)0, acc, /*reuse_a=*/false, /*reuse_b=*/false);
  }

  // D layout: VGPR j, lanes 0-15 -> row M=j, col N=lane. Rows 0..7 / cols 0..1.
  if (lane < 2) {
#pragma unroll
    for (int j = 0; j < 8; ++j) lg[j][lane] = acc[j] + b2[lane];
  }
  __syncthreads();
  if (lane < 16) {
    int mm = lane >> 1, n = lane & 1;
    float l0 = lg[mm][0], l1 = lg[mm][1];
    float mx = fmaxf(l0, l1);
    float e0 = expf(l0 - mx), e1 = expf(l1 - mx);
    out[mm * 2 + n] = ((n == 0) ? e0 : e1) / (e0 + e1);
  }
}

extern "C" void kernel_launch(void* const* d_in, const int* in_sizes, int n_in,
                              void* d_out, int out_size, void* d_ws, size_t ws_size,
                              hipStream_t stream) {
  const float* x  = (const float*)d_in[0];   // (8,2)
  const float* W1 = (const float*)d_in[1];   // (320,2)
  const float* b1 = (const float*)d_in[2];   // (320,)
  const float* qw = (const float*)d_in[3];   // (320,5,10,3)
  const float* W2 = (const float*)d_in[4];   // (2,320)
  const float* b2 = (const float*)d_in[5];   // (2,)
  float* out = (float*)d_out;                // (8,2) softmax
  float* z   = (float*)d_ws;                 // 2560 floats scratch

  qsim_kernel<<<NB * NCIRC, 256, 0, stream>>>(x, W1, b1, qw, z);
  head_kernel<<<1, 32, 0, stream>>>(z, W2, b2, out);
}